// CF_LGCN_E_34222299414919
// MI455X (gfx1250) — compile-verified
//
#include <hip/hip_runtime.h>
#include <cstdint>

// ---------------------------------------------------------------------------
// LightGCN 3-hop propagation (unsorted-edge SpMM via scatter-add atomics).
// gfx1250-specific paths: GLOBAL_LOAD_ASYNC_TO_LDS_B128 (ASYNCcnt-tracked
// async gather, double-buffered per wave), s_wait_asynccnt, global_prefetch_b8,
// native global_atomic_add_f32, NT temporal hints to keep the hot u/i tables
// resident in the 192MB L2 while index/output streams bypass it.
// ---------------------------------------------------------------------------

typedef float v4f __attribute__((ext_vector_type(4)));

#define WAIT_ASYNC(N) asm volatile("s_wait_asynccnt " #N ::: "memory")

__device__ __forceinline__ void async_gather_b128(uint32_t lds_off, uint64_t gaddr) {
  // VDST = LDS byte address (VGPR), VADDR = 64-bit global address, no SADDR.
  asm volatile("global_load_async_to_lds_b128 %0, %1, off"
               :: "v"(lds_off), "v"(gaddr)
               : "memory");
}

// One wave handles 2 edges per step: lanes 0-15 -> edge 2p, lanes 16-31 -> 2p+1.
// Each lane owns one float4 (16B) of the 256B source row.
__global__ __launch_bounds__(256)
void spmm_scatter_async(const float* __restrict__ src,
                        const int*  __restrict__ sidx_arr,
                        const int*  __restrict__ didx_arr,
                        float*      __restrict__ dst,
                        long long E)
{
  __shared__ float4 stage[2][8][32];   // [buf][wave-in-block][lane], 8 KB

  const int lane = threadIdx.x & 31;
  const int wl   = threadIdx.x >> 5;
  const int sub  = lane >> 4;          // which edge of the pair
  const int quad = lane & 15;          // which float4 of the 64-float row

  const long long wglobal = ((long long)blockIdx.x * blockDim.x + threadIdx.x) >> 5;
  const long long nw      = ((long long)gridDim.x * blockDim.x) >> 5;
  const long long npairs  = (E + 1) >> 1;
  if (wglobal >= npairs) return;

  auto issue = [&](int buf, long long p) {
    long long e = 2 * p + sub;
    if (e >= E) e = E - 1;                               // clamp: address-safe
    const int s = __builtin_nontemporal_load(&sidx_arr[e]);   // stream indices past L2
    const uint64_t ga = (uint64_t)(uintptr_t)(src + ((long long)s << 6) + (quad << 2));
    // Low 32 bits of a generic pointer to __shared__ == LDS byte address.
    const uint32_t loff = (uint32_t)(uintptr_t)(void*)&stage[buf][wl][lane];
    async_gather_b128(loff, ga);
  };

  int buf = 0;
  issue(buf, wglobal);                                   // prologue
  for (long long p = wglobal; p < npairs; p += nw) {
    long long pn = p + nw;
    if (pn >= npairs) pn = p;                            // harmless re-issue at tail
    __builtin_prefetch(&sidx_arr[2 * pn], 0, 0);         // global_prefetch_b8
    __builtin_prefetch(&didx_arr[2 * pn], 0, 0);
    issue(buf ^ 1, pn);                                  // keep 2 DMAs in flight
    WAIT_ASYNC(1);                                       // oldest (current buf) done

    const long long e = 2 * p + sub;
    const float4 v = stage[buf][wl][lane];
    if (e < E) {
      const int d = __builtin_nontemporal_load(&didx_arr[e]);
      float* dp = dst + ((long long)d << 6) + (quad << 2);
      __hip_atomic_fetch_add(dp + 0, v.x, __ATOMIC_RELAXED, __HIP_MEMORY_SCOPE_AGENT);
      __hip_atomic_fetch_add(dp + 1, v.y, __ATOMIC_RELAXED, __HIP_MEMORY_SCOPE_AGENT);
      __hip_atomic_fetch_add(dp + 2, v.z, __ATOMIC_RELAXED, __HIP_MEMORY_SCOPE_AGENT);
      __hip_atomic_fetch_add(dp + 3, v.w, __ATOMIC_RELAXED, __HIP_MEMORY_SCOPE_AGENT);
    }
    buf ^= 1;
  }
}

// out_user = 0.25 * (u_x0 + (u_x1 + u_x3))   [u13 accumulated in one buffer]
// Pure streaming pass: NT loads + NT stores, nothing kept in cache.
__global__ __launch_bounds__(256)
void finalize_user(const v4f* __restrict__ u0,
                   const v4f* __restrict__ u13,
                   v4f*       __restrict__ out,
                   long long n4)
{
  const long long stride = (long long)gridDim.x * blockDim.x;
  for (long long i = (long long)blockIdx.x * blockDim.x + threadIdx.x; i < n4; i += stride) {
    const v4f a = __builtin_nontemporal_load(&u0[i]);
    const v4f b = __builtin_nontemporal_load(&u13[i]);
    const v4f r = 0.25f * (a + b);
    __builtin_nontemporal_store(r, &out[i]);
  }
}

// out_item_acc = 0.25 * i_x2 ; out_item0 = item_emb (verbatim)
__global__ __launch_bounds__(256)
void finalize_item(const v4f* __restrict__ i2,
                   const v4f* __restrict__ i0,
                   v4f*       __restrict__ out_acc,
                   v4f*       __restrict__ out_copy,
                   long long n4)
{
  const long long stride = (long long)gridDim.x * blockDim.x;
  for (long long i = (long long)blockIdx.x * blockDim.x + threadIdx.x; i < n4; i += stride) {
    const v4f a = __builtin_nontemporal_load(&i2[i]);
    const v4f c = __builtin_nontemporal_load(&i0[i]);
    __builtin_nontemporal_store(0.25f * a, &out_acc[i]);
    __builtin_nontemporal_store(c, &out_copy[i]);
  }
}

extern "C" void kernel_launch(void* const* d_in, const int* in_sizes, int n_in,
                              void* d_out, int out_size, void* d_ws, size_t ws_size,
                              hipStream_t stream) {
  (void)n_in; (void)out_size; (void)ws_size;
  const float* item_emb = (const float*)d_in[0];     // (I, 64)
  const float* u_x0     = (const float*)d_in[1];     // (U, 64)
  const int*   user_idx = (const int*)d_in[2];       // (E,)
  const int*   item_idx = (const int*)d_in[3];       // (E,)

  const long long I = (long long)in_sizes[0] / 64;
  const long long U = (long long)in_sizes[1] / 64;
  const long long E = (long long)in_sizes[2];

  // Workspace layout: [ u_acc13 : U*64 | i_x2 : I*64 ]  (~77 MB)
  float* u13 = (float*)d_ws;
  float* i2  = u13 + U * 64;
  hipMemsetAsync(d_ws, 0, (size_t)(U + I) * 64 * sizeof(float), stream);

  const int sblocks = 8192, sthreads = 256;
  // hop 1: users += gather(item_emb)   -> u13 (== u_x1 for now)
  spmm_scatter_async<<<sblocks, sthreads, 0, stream>>>(item_emb, item_idx, user_idx, u13, E);
  // hop 2: items += gather(u_x1)       -> i2   (reads u13 before hop 3 writes it)
  spmm_scatter_async<<<sblocks, sthreads, 0, stream>>>(u13, user_idx, item_idx, i2, E);
  // hop 3: users += gather(i_x2)       -> accumulates into u13 (u_x1 + u_x3)
  spmm_scatter_async<<<sblocks, sthreads, 0, stream>>>(i2, item_idx, user_idx, u13, E);

  // Output layout: [ user_acc : U*64 | item_acc : I*64 | item_emb : I*64 ]
  float* out          = (float*)d_out;
  float* out_user     = out;
  float* out_item_acc = out + U * 64;
  float* out_item0    = out_item_acc + I * 64;

  const long long nu4 = U * 16, ni4 = I * 16;
  finalize_user<<<(int)((nu4 + 255) / 256), 256, 0, stream>>>(
      (const v4f*)u_x0, (const v4f*)u13, (v4f*)out_user, nu4);
  finalize_item<<<(int)((ni4 + 255) / 256), 256, 0, stream>>>(
      (const v4f*)i2, (const v4f*)item_emb, (v4f*)out_item_acc, (v4f*)out_item0, ni4);
}